// RelationPredictionNetwork_15204184228370
// MI455X (gfx1250) — compile-verified
//
#include <hip/hip_runtime.h>

typedef __attribute__((ext_vector_type(2))) float v2f;
typedef __attribute__((ext_vector_type(8))) float v8f;
typedef unsigned int u32x4 __attribute__((ext_vector_type(4)));
typedef int          i32x4 __attribute__((ext_vector_type(4)));
typedef int          i32x8 __attribute__((ext_vector_type(8)));

#define NN 512   // nodes
#define DD 1024  // input feature dim
#define HH 512   // hidden dim

#if defined(__HIP_DEVICE_COMPILE__) && __has_builtin(__builtin_amdgcn_tensor_load_to_lds)
#define USE_TDM 1
#else
#define USE_TDM 0
#endif

__device__ __forceinline__ float relu_f(float x) { return x > 0.f ? x : 0.f; }

#if USE_TDM
// --- Tensor Data Mover descriptor helpers (CDNA5 ISA) ----
__device__ __forceinline__ unsigned lds_off32(const void* p) {
    // flat LDS-aperture address: low 32 bits are the LDS byte offset
    return (unsigned)(unsigned long long)p;
}

// Group 0: count=1 [1:0], lds_addr [63:32], global_addr [120:64], type=2 [127:126]
__device__ __forceinline__ u32x4 tdm_g0(unsigned lds_off, const void* g) {
    union { unsigned long long q[2]; u32x4 v; } u;
    u.q[0] = 1ull | ((unsigned long long)lds_off << 32);
    u.q[1] = ((unsigned long long)(size_t)g & 0x01FFFFFFFFFFFFFFull) | (2ull << 62);
    return u.v;
}

// Group 1: data_size[17:16]=2 (4B), pad_en[20], pad_interval[24:22], pad_amount[31:25],
//          tensor_dim0[79:48], tensor_dim1[111:80], tile_dim0[127:112],
//          tile_dim1[143:128], tensor_dim0_stride[207:160]
__device__ __forceinline__ i32x8 tdm_g1(unsigned td0, unsigned td1,
                                        unsigned tile0, unsigned tile1,
                                        unsigned long long stride0,
                                        unsigned pad_en, unsigned pad_int,
                                        unsigned pad_amt) {
    union { unsigned long long q[4]; i32x8 v; } u;
    u.q[0] = (2ull << 16)
           | ((unsigned long long)pad_en  << 20)
           | ((unsigned long long)pad_int << 22)
           | ((unsigned long long)pad_amt << 25)
           | ((unsigned long long)td0     << 48);
    u.q[1] = ((unsigned long long)td0 >> 16)
           | ((unsigned long long)td1 << 16)
           | ((unsigned long long)tile0 << 48);
    u.q[2] = (unsigned long long)tile1
           | ((stride0 & 0xffffffffull) << 32);
    u.q[3] = (stride0 >> 32);
    return u.v;
}

__device__ __forceinline__ void tdm_load(unsigned lds_off, const void* g, i32x8 g1) {
    const i32x4 z4 = {0, 0, 0, 0};
#if defined(__clang_major__) && __clang_major__ >= 23
    const i32x8 z8 = {0, 0, 0, 0, 0, 0, 0, 0};
    __builtin_amdgcn_tensor_load_to_lds(tdm_g0(lds_off, g), g1, z4, z4, z8, 0);
#else
    __builtin_amdgcn_tensor_load_to_lds(tdm_g0(lds_off, g), g1, z4, z4, 0);
#endif
}
#endif // USE_TDM

// ---------------------------------------------------------------------------
// f32 WMMA GEMM: C[M,N] = op(A[M,K]) @ B[K,N]  (+bias) (+Dadd) (relu)
// One wave owns a 16x16 C tile via V_WMMA_F32_16X16X4_F32; 8 waves/block ->
// 32(M) x 64(N) block tile. Tiles staged in LDS by the Tensor Data Mover,
// double-buffered on TENSORcnt; TDM pad inserts 1-2 DWORDs per 64-DWORD row
// (LDS strides 65/66) for banking and 8B-aligned A fragments.
// ---------------------------------------------------------------------------
template<bool BIAS, bool RELU, bool ADD, bool RELUA>
__global__ __launch_bounds__(256) void wmma_gemm_kernel(
    const float* __restrict__ A, const float* __restrict__ B,
    const float* __restrict__ bias, const float* __restrict__ Dadd,
    float* __restrict__ C, int K, int lda, int ldb, int ldc)
{
    const int lane = threadIdx.x & 31;
    const int wave = threadIdx.x >> 5;
    const int wm   = wave >> 2;                 // 0..1
    const int wn   = wave & 3;                  // 0..3
    const int row0 = blockIdx.y * 32 + wm * 16;
    const int col0 = blockIdx.x * 64 + wn * 16;
    const int m    = lane & 15;
    const int kh   = lane >> 4;                 // K-half held by this lane

    v8f acc = {};

#if USE_TDM
    constexpr int KT     = 64;
    constexpr int LDA_SH = KT + 2;              // 66 dwords (pad_amount=1 -> 2 dw)
    constexpr int LDB_SH = 64 + 1;              // 65 dwords (pad_amount=0 -> 1 dw)
    __shared__ float a_sh[2][32 * LDA_SH];
    __shared__ float b_sh[2][KT * LDB_SH];

    const float* Ablk = A + (size_t)(blockIdx.y * 32) * lda;
    const float* Bblk = B + blockIdx.x * 64;

    const i32x8 g1a = tdm_g1((unsigned)lda, 65536u, KT, 32,
                             (unsigned long long)lda, 1u, 5u, 1u);
    const i32x8 g1b = tdm_g1((unsigned)ldb, 65536u, 64, KT,
                             (unsigned long long)ldb, 1u, 5u, 0u);

    const unsigned a_off[2] = { lds_off32(&a_sh[0][0]), lds_off32(&a_sh[1][0]) };
    const unsigned b_off[2] = { lds_off32(&b_sh[0][0]), lds_off32(&b_sh[1][0]) };

    const int  nkt  = K / KT;
    const bool lead = (__builtin_amdgcn_readfirstlane(wave) == 0); // uniform branch:
                                                                   // TDM ignores EXEC,
                                                                   // so must branch-around
    if (lead) {
        tdm_load(a_off[0], Ablk, g1a);
        tdm_load(b_off[0], Bblk, g1b);
    }

    for (int kt = 0; kt < nkt; ++kt) {
        const int cur = kt & 1;
        if (lead) {
            if (kt + 1 < nkt) {
                tdm_load(a_off[cur ^ 1], Ablk + (size_t)(kt + 1) * KT, g1a);
                tdm_load(b_off[cur ^ 1], Bblk + (size_t)(kt + 1) * KT * ldb, g1b);
                __builtin_amdgcn_s_wait_tensorcnt(2);   // in-order: pair kt done
            } else {
                __builtin_amdgcn_s_wait_tensorcnt(0);
            }
        }
        __syncthreads();                                 // tile[cur] visible to all

        const float* alds = &a_sh[cur][m * LDA_SH + kh * 2];
        const float* blds = &b_sh[cur][(kh * 2) * LDB_SH + wn * 16 + m];
        #pragma unroll
        for (int kk = 0; kk < KT; kk += 4) {
            v2f a = *(const v2f*)(alds + kk);            // 8B-aligned ds load
            if (RELUA) { a.x = relu_f(a.x); a.y = relu_f(a.y); }
            v2f b;
            b.x = blds[kk * LDB_SH];
            b.y = blds[kk * LDB_SH + LDB_SH];
            acc = __builtin_amdgcn_wmma_f32_16x16x4_f32(
                      false, a, false, b, (short)0, acc, false, false);
        }
        __syncthreads();                                 // done reading before overwrite
    }
#else
    // Direct-from-global fallback (toolchain without the TDM builtin / host pass)
    const float* ap = A + (size_t)(row0 + m) * lda + kh * 2;
    const float* bp = B + (size_t)(kh * 2) * ldb + col0 + m;
    #pragma unroll 4
    for (int k = 0; k < K; k += 4) {
        v2f a = *(const v2f*)(ap + k);
        if (RELUA) { a.x = relu_f(a.x); a.y = relu_f(a.y); }
        v2f b;
        b.x = bp[(size_t)k * ldb];
        b.y = bp[(size_t)k * ldb + ldb];
        acc = __builtin_amdgcn_wmma_f32_16x16x4_f32(
                  false, a, false, b, (short)0, acc, false, false);
    }
#endif

    // Epilogue: C layout -> VGPR r holds M=r (lanes 0-15) / M=r+8 (lanes 16-31)
    const int col = col0 + m;
    float bv = 0.f;
    if (BIAS) bv = bias[col];
    #pragma unroll
    for (int r = 0; r < 8; ++r) {
        const int row = row0 + r + kh * 8;
        float v = acc[r];
        if (BIAS) v += bv;
        if (ADD)  v += Dadd[(size_t)row * ldc + col];
        if (RELU) v = relu_f(v);
        C[(size_t)row * ldc + col] = v;
    }
}

// ---------------------------------------------------------------------------
// Sequential node scan: h[u] = relu(h[u] + relu(adj[u]) . h), u = 0..511.
// Column c of h evolves independently -> one thread per column, whole column
// staged in LDS (512 rows * 128 cols * 4B = 256KB dynamic LDS, < 320KB/WGP).
// adj row reads are wave-uniform -> scalar loads. No barriers needed at all.
// ---------------------------------------------------------------------------
__global__ __launch_bounds__(128) void scan_kernel(
    const float* __restrict__ adj, float* __restrict__ h)
{
    extern __shared__ float cols[];             // [512][128]
    const int tid = threadIdx.x;
    const int c   = blockIdx.x * 128 + tid;

    for (int v = 0; v < NN; ++v)
        cols[v * 128 + tid] = h[(size_t)v * HH + c];

    for (int u = 0; u < NN; ++u) {
        const float* arow = adj + (size_t)u * NN;
        float s = 0.f;
        #pragma unroll 8
        for (int v = 0; v < NN; ++v) {
            float a = arow[v];
            a = a > 0.f ? a : 0.f;              // A = where(adj>0, adj, 0)
            s = fmaf(a, cols[v * 128 + tid], s);
        }
        cols[u * 128 + tid] = relu_f(cols[u * 128 + tid] + s);
    }

    for (int v = 0; v < NN; ++v)
        h[(size_t)v * HH + c] = cols[v * 128 + tid];
}

// ---------------------------------------------------------------------------
// Fused relation head. One block per u. Never materializes the 512MB e-tensor.
//   logit[u,v] = sum_h relu(a[u,h] + b[v,h] + (EF[uv,:] @ We)[h] + b1[h]) * w2[h]
//   probs[u,v] = sigmoid(logit + b2) * (u != v)
// ---------------------------------------------------------------------------
__global__ __launch_bounds__(256) void relation_kernel(
    const float* __restrict__ amat, const float* __restrict__ bmat,
    const float* __restrict__ ef,   const float* __restrict__ rel_w1,
    const float* __restrict__ rel_b1, const float* __restrict__ rel_w2,
    const float* __restrict__ rel_b2, float* __restrict__ probs)
{
    __shared__ float au[HH];
    __shared__ float w2[HH];
    __shared__ float we[9 * HH];

    const int u   = blockIdx.x;
    const int tid = threadIdx.x;

    for (int i = tid; i < HH; i += 256) {
        au[i] = amat[(size_t)u * HH + i] + rel_b1[i];
        w2[i] = rel_w2[i];
    }
    for (int i = tid; i < 9 * HH; i += 256)
        we[i] = rel_w1[(size_t)(2 * HH) * HH + i];   // We = rel_w1[1024:1033, :]
    __syncthreads();

    const int lane = tid & 31;
    const int wave = tid >> 5;
    const float rb2 = rel_b2[0];

    for (int v = wave; v < NN; v += 8) {
        const float* e = ef + (size_t)(u * NN + v) * 9;
        const float e0 = e[0], e1 = e[1], e2 = e[2], e3 = e[3], e4 = e[4],
                    e5 = e[5], e6 = e[6], e7 = e[7], e8 = e[8];
        const float* brow = bmat + (size_t)v * HH;

        float acc = 0.f;
        #pragma unroll
        for (int hh = 0; hh < HH / 32; ++hh) {
            const int h = lane + hh * 32;
            float t = au[h] + brow[h];
            t = fmaf(e0, we[0 * HH + h], t);
            t = fmaf(e1, we[1 * HH + h], t);
            t = fmaf(e2, we[2 * HH + h], t);
            t = fmaf(e3, we[3 * HH + h], t);
            t = fmaf(e4, we[4 * HH + h], t);
            t = fmaf(e5, we[5 * HH + h], t);
            t = fmaf(e6, we[6 * HH + h], t);
            t = fmaf(e7, we[7 * HH + h], t);
            t = fmaf(e8, we[8 * HH + h], t);
            t = relu_f(t);
            acc = fmaf(t, w2[h], acc);
        }
        #pragma unroll
        for (int off = 16; off > 0; off >>= 1)
            acc += __shfl_xor(acc, off, 32);

        if (lane == 0) {
            float p = 1.f / (1.f + __expf(-(acc + rb2)));
            if (v == u) p = 0.f;
            probs[(size_t)u * NN + v] = p;
        }
    }
}

// ---------------------------------------------------------------------------
extern "C" void kernel_launch(void* const* d_in, const int* in_sizes, int n_in,
                              void* d_out, int out_size, void* d_ws, size_t ws_size,
                              hipStream_t stream)
{
    const float* node_features = (const float*)d_in[0];   // (512,1024)
    const float* adjacency     = (const float*)d_in[1];   // (512,512)
    const float* edge_features = (const float*)d_in[2];   // (512*512, 9)
    const float* gnn1_w        = (const float*)d_in[3];   // (1024,512)
    const float* gnn1_b        = (const float*)d_in[4];   // (512,)
    const float* gnn2_w        = (const float*)d_in[5];   // (512,512)
    const float* gnn2_b        = (const float*)d_in[6];   // (512,)
    const float* rel_w1        = (const float*)d_in[7];   // (1033,512)
    const float* rel_b1        = (const float*)d_in[8];   // (512,)
    const float* rel_w2        = (const float*)d_in[9];   // (512,1)
    const float* rel_b2        = (const float*)d_in[10];  // (1,)

    float* out     = (float*)d_out;
    float* h_final = out;                 // first 512*512 outputs
    float* probs   = out + NN * HH;       // second 512*512 outputs

    float* ws   = (float*)d_ws;
    float* h1   = ws;                     // 512*512
    float* h2   = ws + 1 * NN * HH;       // 512*512
    float* amat = ws + 2 * NN * HH;       // 512*512
    float* bmat = ws + 3 * NN * HH;       // 512*512

    const dim3 blk(256);
    const dim3 grid(HH / 64, NN / 32);    // N-tiles x M-tiles (512x512 output)

    // 1) h1 = relu(NF @ gnn1_w + b1)          [512x1024x512]
    wmma_gemm_kernel<true, true, false, false><<<grid, blk, 0, stream>>>(
        node_features, gnn1_w, gnn1_b, nullptr, h1, DD, DD, HH, HH);

    // 2) sequential scan, in place on h1      [per-column recurrences in LDS]
    scan_kernel<<<dim3(HH / 128), dim3(128), NN * 128 * sizeof(float), stream>>>(
        adjacency, h1);

    // 3) h2 = relu(h1 @ gnn2_w + b2)          [512^3]
    wmma_gemm_kernel<true, true, false, false><<<grid, blk, 0, stream>>>(
        h1, gnn2_w, gnn2_b, nullptr, h2, HH, HH, HH, HH);

    // 4) h_final = relu(h2 + relu(adj) @ h2)  [512^3, relu applied to A on load]
    wmma_gemm_kernel<false, true, true, true><<<grid, blk, 0, stream>>>(
        adjacency, h2, nullptr, h2, h_final, NN, NN, HH, HH);

    // 5) a = h_final @ Wu ; b = h_final @ Wv  [2 x 512^3]
    wmma_gemm_kernel<false, false, false, false><<<grid, blk, 0, stream>>>(
        h_final, rel_w1, nullptr, nullptr, amat, HH, HH, HH, HH);
    wmma_gemm_kernel<false, false, false, false><<<grid, blk, 0, stream>>>(
        h_final, rel_w1 + (size_t)HH * HH, nullptr, nullptr, bmat, HH, HH, HH, HH);

    // 6) fused relation head -> probs
    relation_kernel<<<dim3(NN), dim3(256), 0, stream>>>(
        amat, bmat, edge_features, rel_w1, rel_b1, rel_w2, rel_b2, probs);
}